// Cooc_Layer_24464133718822
// MI455X (gfx1250) — compile-verified
//
#include <hip/hip_runtime.h>
#include <hip/hip_bf16.h>

typedef __attribute__((ext_vector_type(2))) float v2f;
typedef __attribute__((ext_vector_type(8))) float v8f;

#define NB    8
#define CIN   512
#define COUT  128
#define HW1   256   // 16*16
#define HW2   196   // 14*14
#define CC    16384 // 128*128
#define SROW2 404   // doubled circulant row: indices j = n - k + 196 in [1, 403]

// ---------------- zero-init the per-(b,k) max buffer ----------------
__global__ void zero_kernel(unsigned int* __restrict__ p, int n) {
    int i = blockIdx.x * blockDim.x + threadIdx.x;
    if (i < n) p[i] = 0u;   // 0u == bits of +0.0f, valid lower bound (all R >= 0)
}

// ---------------- GEMM1: y[b,m,n] = relu(sum_k w[m,k] * x[b,k,n]) ----------------
// M=128, K=512, N=256 per batch. 4 waves/block, 1 wave = one 16x16 D tile.
__global__ __launch_bounds__(128) void gemm1_kernel(const float* __restrict__ x,
                                                    const float* __restrict__ w,
                                                    float* __restrict__ y) {
    const int wave = threadIdx.x >> 5;
    const int lane = threadIdx.x & 31;
    const int b    = blockIdx.x >> 5;                 // 32 blocks per batch
    const int tile = ((blockIdx.x & 31) << 2) + wave; // 0..127
    const int mt   = tile >> 4;                       // 0..7
    const int nt   = tile & 15;                       // 0..15
    const int koff = (lane >> 4) << 1;                // 0 or 2
    const int nh   = lane & 15;
    const int m    = mt * 16 + nh;
    const int n    = nt * 16 + nh;
    const float* __restrict__ wrow = w + (size_t)m * CIN + koff;
    const float* __restrict__ xcol = x + (size_t)b * CIN * HW1 + (size_t)koff * HW1 + n;

    v8f acc = {};
    for (int k0 = 0; k0 < CIN; k0 += 4) {
        v2f a, bf;
        a.x  = wrow[k0];
        a.y  = wrow[k0 + 1];
        bf.x = xcol[k0 * HW1];
        bf.y = xcol[k0 * HW1 + HW1];
        acc = __builtin_amdgcn_wmma_f32_16x16x4_f32(false, a, false, bf,
                                                    (short)0, acc, false, false);
    }

    float* __restrict__ yb = y + (size_t)b * COUT * HW1;
    const int mbase = mt * 16 + ((lane >> 4) << 3);
#pragma unroll
    for (int r = 0; r < 8; ++r) {
        float v = acc[r];
        v = v > 0.0f ? v : 0.0f;                      // ReLU
        yb[(mbase + r) * HW1 + n] = v;
    }
}

// ---------------- depthwise 3x3 Gaussian, VALID: (16x16) -> (14x14) ----------------
__global__ void dwconv_kernel(const float* __restrict__ y, float* __restrict__ xf) {
    int i = blockIdx.x * blockDim.x + threadIdx.x;
    if (i >= NB * COUT * HW2) return;
    int p  = i % HW2;
    int bc = i / HW2;
    int ow = p % 14, oh = p / 14;
    const float* __restrict__ src = y + (size_t)bc * HW1;
    const float g0 = 0.63661977236758138f;                  // 1/(2*pi*0.25)
    const float g1 = g0 * 0.13533528323661270f;             // * exp(-2)
    const float g2 = g0 * 0.018315638888734179f;            // * exp(-4)
    float a = 0.0f;
    a += g2 * src[(oh + 0) * 16 + ow + 0] + g1 * src[(oh + 0) * 16 + ow + 1] + g2 * src[(oh + 0) * 16 + ow + 2];
    a += g1 * src[(oh + 1) * 16 + ow + 0] + g0 * src[(oh + 1) * 16 + ow + 1] + g1 * src[(oh + 1) * 16 + ow + 2];
    a += g2 * src[(oh + 2) * 16 + ow + 0] + g1 * src[(oh + 2) * 16 + ow + 1] + g2 * src[(oh + 2) * 16 + ow + 2];
    xf[i] = a;
}

// ---------------- correlation GEMM + fused flat-strided max ----------------
// Block = (b, c2). R_block = A(128x196) x S(196x196), S[k,p] = srow[(p-k) mod 196].
// LDS holds the doubled row srow2[j] = srow[j % 196] so S[k,n] = srow2[n - k + 196]
// with no per-iteration modulo. Padded columns (n >= 196) compute garbage-but-finite
// values that are masked at the store, so no in-loop select is needed.
// Each stored R element folds into slot k = (a*25088 + c2*196 + p) & 16383 via
// uint atomicMax (valid: all values are non-negative floats).
__global__ __launch_bounds__(128) void corr_kernel(const float* __restrict__ xf,
                                                   unsigned int* __restrict__ maxbuf) {
    __shared__ float srow2[SROW2];
    const int b  = blockIdx.x >> 7;
    const int c2 = blockIdx.x & 127;
    const float* __restrict__ xfb = xf + (size_t)b * COUT * HW2;
    for (int i = threadIdx.x; i < SROW2; i += blockDim.x) {
        int j = i;
        if (j >= 2 * HW2) j -= 2 * HW2;
        else if (j >= HW2) j -= HW2;
        srow2[i] = xfb[c2 * HW2 + j];
    }
    __syncthreads();

    const int wave = threadIdx.x >> 5;
    const int lane = threadIdx.x & 31;
    const int koff = (lane >> 4) << 1;                // 0 or 2
    const int nh   = lane & 15;
    unsigned int* __restrict__ mb = maxbuf + b * CC;

    for (int t = wave; t < 104; t += 4) {             // 8 M-tiles x 13 N-tiles
        const int mt = t / 13;
        const int nt = t % 13;
        const int m  = mt * 16 + nh;
        const int n  = nt * 16 + nh;                  // may reach 207 (padded)
        const float* __restrict__ arow = xfb + (size_t)m * HW2 + koff;
        const float* __restrict__ sp   = srow2 + n + HW2 - koff;
        v8f acc = {};
        for (int k0 = 0; k0 < HW2; k0 += 4) {         // 49 WMMA steps, wave-uniform
            v2f a, bf;
            a.x  = arow[k0];
            a.y  = arow[k0 + 1];
            bf.x = sp[-k0];                           // srow2[n - (k0+koff) + 196]
            bf.y = sp[-k0 - 1];
            acc = __builtin_amdgcn_wmma_f32_16x16x4_f32(false, a, false, bf,
                                                        (short)0, acc, false, false);
        }
        if (n < HW2) {                                // divergence only after the WMMAs
            const int abase = mt * 16 + ((lane >> 4) << 3);
#pragma unroll
            for (int r = 0; r < 8; ++r) {
                const int f = (abase + r) * 25088 + c2 * HW2 + n;
                atomicMax(&mb[f & (CC - 1)], __float_as_uint(acc[r]));
            }
        }
    }
}

// ---------------- finalize: out[b,k] = sqrt(max) / (sum_k max + eps) ----------------
__global__ __launch_bounds__(256) void finalize_kernel(const unsigned int* __restrict__ maxbuf,
                                                       float* __restrict__ out) {
    __shared__ float red[256];
    const int b = blockIdx.x;
    const unsigned int* __restrict__ mb = maxbuf + b * CC;
    float s = 0.0f;
    for (int k = threadIdx.x; k < CC; k += 256) s += __uint_as_float(mb[k]); // sum of c_ij^2
    red[threadIdx.x] = s;
    __syncthreads();
    for (int off = 128; off > 0; off >>= 1) {
        if (threadIdx.x < off) red[threadIdx.x] += red[threadIdx.x + off];
        __syncthreads();
    }
    const float norm = red[0] + 1e-11f;
    for (int k = threadIdx.x; k < CC; k += 256)
        out[b * CC + k] = sqrtf(__uint_as_float(mb[k])) / norm;
}

extern "C" void kernel_launch(void* const* d_in, const int* in_sizes, int n_in,
                              void* d_out, int out_size, void* d_ws, size_t ws_size,
                              hipStream_t stream) {
    const float* x = (const float*)d_in[0];   // (8,512,16,16)
    const float* w = (const float*)d_in[1];   // (128,512)
    float* out = (float*)d_out;               // (8,16384)

    char* ws = (char*)d_ws;
    float* y  = (float*)ws;                                   // 8*128*256*4 = 1,048,576 B
    float* xf = (float*)(ws + 1048576);                       // 8*128*196*4 =   802,816 B
    unsigned int* maxbuf = (unsigned int*)(ws + 1048576 + 802816); // 8*16384*4 = 524,288 B

    zero_kernel<<<(NB * CC + 255) / 256, 256, 0, stream>>>(maxbuf, NB * CC);
    gemm1_kernel<<<NB * 32, 128, 0, stream>>>(x, w, y);
    dwconv_kernel<<<(NB * COUT * HW2 + 255) / 256, 256, 0, stream>>>(y, xf);
    corr_kernel<<<NB * COUT, 128, 0, stream>>>(xf, maxbuf);
    finalize_kernel<<<NB, 256, 0, stream>>>(maxbuf, out);
}